// SignMagnitudeLoss_13804024889845
// MI455X (gfx1250) — compile-verified
//
#include <hip/hip_runtime.h>
#include <math.h>

typedef __attribute__((ext_vector_type(2))) float v2f;
typedef __attribute__((ext_vector_type(8))) float v8f;

#define B_    4
#define H0    544
#define W0    960
#define HH    272          // half-res height
#define WH    480          // half-res width
#define MDISP 24           // max disparity at half res
#define WEXT  (WH + 2*MDISP)   // 528, extended right-stats domain
#define ND    49           // d_idx 0..48  (d = d_idx - 24; d_idx==24 skipped)
#define TY    10           // output rows per block
#define ROWS  (TY + 10)    // 20 rows of horizontal cross sums buffered
#define INV121 (1.0f/121.0f)

// ---------------- small kernels ----------------

__global__ void init_acc_k(float* acc) {
    if (threadIdx.x < 4) acc[threadIdx.x] = 0.0f;
}

// 2x2 avg-pool + channel mean: out[b][y][x] = (sum over 3ch of 2x2 block)/12
__global__ void downsample_k(const float* __restrict__ img, float* __restrict__ out) {
    int idx = blockIdx.x * blockDim.x + threadIdx.x;
    int total = B_ * HH * WH;
    if (idx >= total) return;
    int x = idx % WH;
    int y = (idx / WH) % HH;
    int b = idx / (WH * HH);
    const float* base = img + (size_t)b * 3 * (size_t)(H0 * W0);
    int Y = 2 * y, X = 2 * x;
    float s = 0.0f;
#pragma unroll
    for (int c = 0; c < 3; ++c) {
        const float* p = base + (size_t)c * (H0 * W0) + (size_t)Y * W0 + X;
        s += p[0] + p[1] + p[W0] + p[W0 + 1];
    }
    out[idx] = s * (1.0f / 12.0f);
}

// horizontal 11-tap sums of g and g^2 at centers xo+xoff (zero outside [0,WH))
__global__ void hpass_k(const float* __restrict__ g, float* __restrict__ h,
                        float* __restrict__ h2, int Wout, int xoff) {
    int idx = blockIdx.x * blockDim.x + threadIdx.x;
    int total = B_ * HH * Wout;
    if (idx >= total) return;
    int xo = idx % Wout;
    int y  = (idx / Wout) % HH;
    int b  = idx / (Wout * HH);
    int xc = xo + xoff;
    const float* row = g + (size_t)(b * HH + y) * WH;
    float s = 0.0f, s2 = 0.0f;
#pragma unroll
    for (int j = -5; j <= 5; ++j) {
        int xx = xc + j;
        int xcl = min(max(xx, 0), WH - 1);
        float msk = (xx >= 0 && xx < WH) ? 1.0f : 0.0f;
        float v = row[xcl] * msk;
        s += v; s2 += v * v;
    }
    h[idx]  = s;
    h2[idx] = s2;
}

// vertical 11-tap sum, produce mean and std (count_include_pad: /121, var floor 1e-8)
__global__ void vpass_k(const float* __restrict__ h, const float* __restrict__ h2,
                        float* __restrict__ om, float* __restrict__ ostd, int Wout) {
    int idx = blockIdx.x * blockDim.x + threadIdx.x;
    int total = B_ * HH * Wout;
    if (idx >= total) return;
    int xo = idx % Wout;
    int y  = (idx / Wout) % HH;
    int b  = idx / (Wout * HH);
    float s = 0.0f, s2 = 0.0f;
#pragma unroll
    for (int i = -5; i <= 5; ++i) {
        int yy = y + i;
        int ycl = min(max(yy, 0), HH - 1);
        float msk = (yy >= 0 && yy < HH) ? 1.0f : 0.0f;
        size_t o = (size_t)(b * HH + ycl) * Wout + xo;
        s += h[o] * msk; s2 += h2[o] * msk;
    }
    float m   = s * INV121;
    float var = fmaxf(s2 * INV121 - m * m, 1e-8f);
    om[idx]   = m;
    ostd[idx] = sqrtf(var);
}

// ---------------- main NCC + loss kernel (WMMA fp32 16x16x4) ----------------
// Per row: C[x, x'] = sum_j lg[x-5+j] * rg[x'-5+j]  (K=11 padded to 12, 3 WMMA steps)
// x in [x0, x0+15], x' in [x0-24, x0+39] -> 4 N-tiles; d_idx = 16n + c - m in [0,48].

__global__ __launch_bounds__(256) void ncc_k(
    const float* __restrict__ lg,  const float* __restrict__ rg,
    const float* __restrict__ lm,  const float* __restrict__ lstd,
    const float* __restrict__ rme, const float* __restrict__ rse,
    const float* __restrict__ pred, float* __restrict__ acc) {

    __shared__ float Hc[ROWS * 16 * ND];   // 15680 floats = 62720 bytes
    __shared__ float junk[32];             // sink for out-of-band scatter writes

    int tid  = threadIdx.x;
    int lane = tid & 31;
    int wave = tid >> 5;
    int m16  = lane & 15;
    int hi   = lane >> 4;

    int x0 = blockIdx.x * 16;
    int y0 = blockIdx.y * TY;
    int b  = blockIdx.z;

    const float* lgb = lg + (size_t)b * HH * WH;
    const float* rgb = rg + (size_t)b * HH * WH;

    // ---- Phase A: horizontal cross sums via WMMA, one row per wave-iteration ----
    for (int r = wave; r < ROWS; r += 8) {
        int gy = y0 - 5 + r;
        float* hrow = &Hc[r * 16 * ND];
        if (gy < 0 || gy >= HH) {
            for (int i = lane; i < 16 * ND; i += 32) hrow[i] = 0.0f;
            continue;
        }
        const float* lrow = lgb + (size_t)gy * WH;
        const float* rrow = rgb + (size_t)gy * WH;

        // A fragments: A[m][k] = lg[x0+m-5+k], lane=(k>=2)*16+m, elem=k%2, k=4s+2*hi+e
        // Unconditional clamped loads + mask select: no EXEC branching, loads batch.
        v2f afrag[3];
#pragma unroll
        for (int s = 0; s < 3; ++s) {
#pragma unroll
            for (int e = 0; e < 2; ++e) {
                int k   = 4 * s + 2 * hi + e;
                int xg  = x0 + m16 - 5 + k;
                int xc  = min(max(xg, 0), WH - 1);
                float msk = (k < 11 && xg >= 0 && xg < WH) ? 1.0f : 0.0f;
                afrag[s][e] = lrow[xc] * msk;
            }
        }

        // All B fragments up front so loads cluster into one wait
        v2f bfrag[4][3];
#pragma unroll
        for (int n = 0; n < 4; ++n) {
#pragma unroll
            for (int s = 0; s < 3; ++s) {
#pragma unroll
                for (int e = 0; e < 2; ++e) {
                    int k   = 4 * s + 2 * hi + e;
                    int xg  = x0 - 24 + 16 * n + m16 - 5 + k;
                    int xc  = min(max(xg, 0), WH - 1);
                    float msk = (k < 11 && xg >= 0 && xg < WH) ? 1.0f : 0.0f;
                    bfrag[n][s][e] = rrow[xc] * msk;
                }
            }
        }

#pragma unroll
        for (int n = 0; n < 4; ++n) {
            v8f c = {};
#pragma unroll
            for (int s = 0; s < 3; ++s) {
                c = __builtin_amdgcn_wmma_f32_16x16x4_f32(
                        false, afrag[s], false, bfrag[n][s], (short)0, c, false, false);
            }
            // scatter band: lane holds column c=m16, rows m = v + 8*hi.
            // Out-of-band elements go to a junk slot via address select (no EXEC branching).
#pragma unroll
            for (int v = 0; v < 8; ++v) {
                int mrow = v + 8 * hi;             // x_local
                int didx = 16 * n + m16 - mrow;    // d + 24
                bool inband = (didx >= 0) && (didx < ND);
                float* dst = inband ? &hrow[mrow * ND + didx] : &junk[lane];
                *dst = c[v];
            }
        }
    }
    __syncthreads();

    // ---- Phase B: in-place inclusive prefix over rows per (x,d) column ----
    for (int col = tid; col < 16 * ND; col += 256) {
        float s = 0.0f;
#pragma unroll 4
        for (int r = 0; r < ROWS; ++r) {
            int o = r * 16 * ND + col;
            s += Hc[o];
            Hc[o] = s;
        }
    }
    __syncthreads();

    // ---- Phase C: per-pixel argmax over d, fused loss partials (branchless) ----
    float sv = 0.0f, mg = 0.0f, ct = 0.0f;
    for (int p = tid; p < TY * 16; p += 256) {
        int t  = p >> 4;
        int xl = p & 15;
        int gy = y0 + t;
        int gx = x0 + xl;
        if (gy >= HH) continue;

        size_t pix  = (size_t)(b * HH + gy) * WH + gx;
        float lmv = lm[pix];
        float lsv = lstd[pix];
        size_t rbase = (size_t)(b * HH + gy) * WEXT + gx;  // + didx gives ext index gx+d+24

        float best = -1.0f;
        int bestd = 0;
        int rowHi = ((t + 10) * 16 + xl) * ND;
        int rowLo = ((t - 1) * 16 + xl) * ND;
        float loMask = (t > 0) ? 1.0f : 0.0f;
        int rowLoSafe = (t > 0) ? rowLo : rowHi;
#pragma unroll 1
        for (int didx = 0; didx < ND; ++didx) {
            if (didx == 24) continue;   // d == 0 excluded; order matches dvals
            float cr = Hc[rowHi + didx] - loMask * Hc[rowLoSafe + didx];
            cr *= INV121;
            float rmv = rme[rbase + didx];
            float rsv = rse[rbase + didx];
            float ncc = (cr - lmv * rmv) / (lsv * rsv + 1e-8f);
            if (ncc > best) { best = ncc; bestd = didx - 24; }
        }

        float nd  = (float)bestd * 2.0f;        // disparity at full res
        float nc  = fmaxf(best, 0.0f);
        float act = (nc > 0.3f) ? 1.0f : 0.0f;  // conf mask & active
        // sign_map = sign(nd) * active; conf masked by active
        float sgn = ((nd > 0.0f) ? 1.0f : ((nd < 0.0f) ? -1.0f : 0.0f)) * act;
        float ncm = nc * act;
        const float* pp = pred + (size_t)b * (H0 * W0) + (size_t)(2 * gy) * W0 + 2 * gx;
#pragma unroll
        for (int iy = 0; iy < 2; ++iy) {
#pragma unroll
            for (int ix = 0; ix < 2; ++ix) {
                float pv = pp[(size_t)iy * W0 + ix];
                sv += fmaxf(-pv * sgn, 0.0f) * act;
                mg += ncm * fabsf(pv - nd);
            }
        }
        ct += 4.0f * act;
    }
    __syncthreads();

    // ---- block reduction (reuse LDS) + global atomics ----
    float* red = Hc;
    red[tid] = sv; red[256 + tid] = mg; red[512 + tid] = ct;
    __syncthreads();
    for (int off = 128; off > 0; off >>= 1) {
        if (tid < off) {
            red[tid]       += red[tid + off];
            red[256 + tid] += red[256 + tid + off];
            red[512 + tid] += red[512 + tid + off];
        }
        __syncthreads();
    }
    if (tid == 0) {
        atomicAdd(&acc[0], red[0]);
        atomicAdd(&acc[1], red[256]);
        atomicAdd(&acc[2], red[512]);
    }
}

__global__ void finalize_k(const float* __restrict__ acc, float* __restrict__ out) {
    float n = fmaxf(acc[2], 1.0f);
    out[0] = (0.3f * acc[0] + 0.7f * acc[1]) / n;
}

// ---------------- launcher ----------------

extern "C" void kernel_launch(void* const* d_in, const int* in_sizes, int n_in,
                              void* d_out, int out_size, void* d_ws, size_t ws_size,
                              hipStream_t stream) {
    const float* pred  = (const float*)d_in[0];
    const float* left  = (const float*)d_in[1];
    const float* right = (const float*)d_in[2];
    float* out = (float*)d_out;

    const size_t NL = (size_t)B_ * HH * WH;    // 522240
    const size_t NE = (size_t)B_ * HH * WEXT;  // 574464

    float* ws  = (float*)d_ws;
    float* acc = ws;            // 4 floats
    float* lg  = ws + 4;
    float* rg  = lg + NL;
    float* lm  = rg + NL;
    float* ls  = lm + NL;
    float* rme = ls + NL;
    float* rse = rme + NE;
    float* t1  = rse + NE;      // temp h-sums (sized NE, reused for left & right)
    float* t2  = t1 + NE;
    // total: 4 + 4*NL + 4*NE floats ~= 17.6 MB

    init_acc_k<<<1, 32, 0, stream>>>(acc);

    int blkNL = (int)((NL + 255) / 256);
    downsample_k<<<blkNL, 256, 0, stream>>>(left,  lg);
    downsample_k<<<blkNL, 256, 0, stream>>>(right, rg);

    // left stats on [0,WH)
    hpass_k<<<blkNL, 256, 0, stream>>>(lg, t1, t2, WH, 0);
    vpass_k<<<blkNL, 256, 0, stream>>>(t1, t2, lm, ls, WH);

    // right stats on extended [-24, WH+24)
    int blkNE = (int)((NE + 255) / 256);
    hpass_k<<<blkNE, 256, 0, stream>>>(rg, t1, t2, WEXT, -MDISP);
    vpass_k<<<blkNE, 256, 0, stream>>>(t1, t2, rme, rse, WEXT);

    dim3 grid(WH / 16, (HH + TY - 1) / TY, B_);   // (30, 28, 4)
    ncc_k<<<grid, 256, 0, stream>>>(lg, rg, lm, ls, rme, rse, pred, acc);

    finalize_k<<<1, 1, 0, stream>>>(acc, out);
}